// EGNNLayer_72138270704321
// MI455X (gfx1250) — compile-verified
//
#include <hip/hip_runtime.h>
#include <math.h>

// EGNN layer on gfx1250: per-node WMMA-f32 message GEMM with hi/bias/u2 folded
// into the K dimension, LDS reductions, and FUSED pass-through copies for
// hj/xj (coalesced b128, L2-hit reads) instead of serialized hipMemcpyAsync.

typedef __attribute__((ext_vector_type(2))) float v2f;
typedef __attribute__((ext_vector_type(8))) float v8f;

#define B_NODES 4096
#define N_NBR   256
#define D_H     64
#define D_X     3
#define D_M     32

__device__ __forceinline__ float silu_f(float x) {
    return x / (1.0f + __expf(-x));
}

__global__ __launch_bounds__(256, 2)
void egnn_wmma_kernel(const float* __restrict__ hi,   const float* __restrict__ xi,
                      const float* __restrict__ hj,   const float* __restrict__ xj,
                      const float* __restrict__ We_w, const float* __restrict__ We_b,
                      const float* __restrict__ Wx_w, const float* __restrict__ Wx_b,
                      const float* __restrict__ Wh_w, const float* __restrict__ Wh_b,
                      float* __restrict__ out_hi, float* __restrict__ out_xi,
                      float* __restrict__ out_hj, float* __restrict__ out_xj)
{
    __shared__ float s_hi[D_H];
    __shared__ float s_ai[D_M];      // hi @ We_w[0:64] + We_b  (per node)
    __shared__ float s_mi[D_M];      // sum over neighbors of mij
    __shared__ float s_wx[D_M];      // Wx_w
    __shared__ float s_xi[D_X];
    __shared__ float s_xacc[D_X];    // sum of w * u_diff
    __shared__ float s_stage[8][16 * D_M];   // per-wave mij tile staging

    const int b      = blockIdx.x;
    const int tid    = threadIdx.x;
    const int wave   = tid >> 5;
    const int lid    = tid & 31;
    const int nlo    = lid & 15;
    const int hihalf = lid >> 4;     // 0: lanes 0-15, 1: lanes 16-31
    const int koff   = hihalf * 2;   // A/B operand K sub-offset for 16x16x4 f32

    if (tid < D_H) s_hi[tid] = hi[(size_t)b * D_H + tid];
    if (tid < D_M) { s_mi[tid] = 0.0f; s_wx[tid] = Wx_w[tid]; }
    if (tid < D_X) { s_xacc[tid] = 0.0f; s_xi[tid] = xi[(size_t)b * D_X + tid]; }
    __syncthreads();

    // a_i[c] = hi . We_w[0:64, c] + We_b[c]   (shared across all 256 neighbors)
    if (tid < D_M) {
        float acc = We_b[tid];
        #pragma unroll 8
        for (int k = 0; k < D_H; ++k) acc += s_hi[k] * We_w[k * D_M + tid];
        s_ai[tid] = acc;
    }
    __syncthreads();

    // ---- Preload B operands for the WMMA chain (K' = 0..67) ----
    // K' 0..63  -> We_w rows 64..127 (hj part)
    // K' 64..66 -> We_w rows 128..130 (u_diff^2 part)
    // K' 67     -> a_i (multiplied by constant 1.0 in A)
    v2f Br0[17], Br1[17];
    #pragma unroll
    for (int kk = 0; kk < 16; ++kk) {
        const int r = D_H + 4 * kk + koff;
        v2f t0, t1;
        t0.x = We_w[(r    ) * D_M + nlo];
        t0.y = We_w[(r + 1) * D_M + nlo];
        t1.x = We_w[(r    ) * D_M + 16 + nlo];
        t1.y = We_w[(r + 1) * D_M + 16 + nlo];
        Br0[kk] = t0; Br1[kk] = t1;
    }
    {
        v2f t0, t1;
        if (hihalf == 0) {
            t0.x = We_w[128 * D_M + nlo];      t0.y = We_w[129 * D_M + nlo];
            t1.x = We_w[128 * D_M + 16 + nlo]; t1.y = We_w[129 * D_M + 16 + nlo];
        } else {
            t0.x = We_w[130 * D_M + nlo];      t0.y = s_ai[nlo];
            t1.x = We_w[130 * D_M + 16 + nlo]; t1.y = s_ai[16 + nlo];
        }
        Br0[16] = t0; Br1[16] = t1;
    }

    const float* hj_b = hj + (size_t)b * N_NBR * D_H;
    const float* xj_b = xj + (size_t)b * N_NBR * D_X;
    const float  wxb  = Wx_b[0];
    float* st = &s_stage[wave][0];

    // ---- 16 neighbor-tiles of 16 rows; 2 tiles per wave ----
    for (int t = wave; t < 16; t += 8) {
        const int nb = t * 16 + nlo;

        // u_diff for this lane's neighbor row (both half-waves compute same row)
        const float d0 = s_xi[0] - xj_b[nb * 3 + 0];
        const float d1 = s_xi[1] - xj_b[nb * 3 + 1];
        const float d2 = s_xi[2] - xj_b[nb * 3 + 2];

        // prefetch next tile's hj row into cache
        if (t + 8 < 16)
            __builtin_prefetch(hj_b + (size_t)((t + 8) * 16 + nlo) * D_H + koff, 0, 1);

        v8f C0 = {0.f,0.f,0.f,0.f,0.f,0.f,0.f,0.f};
        v8f C1 = {0.f,0.f,0.f,0.f,0.f,0.f,0.f,0.f};

        const float* arow = hj_b + (size_t)nb * D_H + koff;
        #pragma unroll
        for (int kk = 0; kk < 16; ++kk) {
            v2f A = *(const v2f*)(arow + 4 * kk);
            C0 = __builtin_amdgcn_wmma_f32_16x16x4_f32(false, A, false, Br0[kk],
                                                       (short)0, C0, false, false);
            C1 = __builtin_amdgcn_wmma_f32_16x16x4_f32(false, A, false, Br1[kk],
                                                       (short)0, C1, false, false);
        }
        // last K step: A columns are [u0,u1 | u2, 1.0]
        v2f Al;
        Al.x = hihalf ? (d2 * d2) : (d0 * d0);
        Al.y = hihalf ? 1.0f      : (d1 * d1);
        C0 = __builtin_amdgcn_wmma_f32_16x16x4_f32(false, Al, false, Br0[16],
                                                   (short)0, C0, false, false);
        C1 = __builtin_amdgcn_wmma_f32_16x16x4_f32(false, Al, false, Br1[16],
                                                   (short)0, C1, false, false);

        // SiLU -> mij; accumulate mi; stage tile to wave-private LDS
        float mi0 = 0.0f, mi1 = 0.0f;
        #pragma unroll
        for (int j = 0; j < 8; ++j) {
            const int mrow = j + hihalf * 8;
            const float x0 = silu_f(C0[j]);
            const float x1 = silu_f(C1[j]);
            mi0 += x0; mi1 += x1;
            st[mrow * D_M + nlo]      = x0;
            st[mrow * D_M + 16 + nlo] = x1;
        }
        atomicAdd(&s_mi[nlo],      mi0);
        atomicAdd(&s_mi[16 + nlo], mi1);

        __builtin_amdgcn_wave_barrier();   // LDS ops of one wave execute in order

        // per-neighbor w = silu(mij . Wx + b); accumulate w * u_diff
        float a0 = 0.0f, a1 = 0.0f, a2 = 0.0f;
        if (lid < 16) {
            float acc = wxb;
            #pragma unroll 8
            for (int c = 0; c < D_M; ++c) acc += st[nlo * D_M + c] * s_wx[c];
            const float w = silu_f(acc);
            a0 = d0 * w; a1 = d1 * w; a2 = d2 * w;
        }
        #pragma unroll
        for (int off = 16; off > 0; off >>= 1) {
            a0 += __shfl_xor(a0, off, 32);
            a1 += __shfl_xor(a1, off, 32);
            a2 += __shfl_xor(a2, off, 32);
        }
        if (lid == 0) {
            atomicAdd(&s_xacc[0], a0);
            atomicAdd(&s_xacc[1], a1);
            atomicAdd(&s_xacc[2], a2);
        }
    }

    // ---- Fused pass-through copy (reads hit L2 after compute touched them) ----
    {
        const float4* __restrict__ s4 = (const float4*)hj_b;
        float4* __restrict__ d4 = (float4*)(out_hj + (size_t)b * N_NBR * D_H);
        #pragma unroll 4
        for (int i = tid; i < (N_NBR * D_H) / 4; i += 256) d4[i] = s4[i];

        const float4* __restrict__ sx = (const float4*)xj_b;
        float4* __restrict__ dx = (float4*)(out_xj + (size_t)b * N_NBR * D_X);
        for (int i = tid; i < (N_NBR * D_X) / 4; i += 256) dx[i] = sx[i];
    }

    __syncthreads();

    // xi_new = xi + mean_n(w * u_diff)
    if (tid < D_X)
        out_xi[(size_t)b * D_X + tid] = s_xi[tid] + s_xacc[tid] * (1.0f / (float)N_NBR);

    // hi_new = silu([hi | mi] @ Wh + b)
    if (tid < D_H) {
        float acc = Wh_b[tid];
        #pragma unroll 8
        for (int k = 0; k < D_H; ++k) acc += s_hi[k] * Wh_w[k * D_H + tid];
        #pragma unroll 8
        for (int k = 0; k < D_M; ++k) acc += s_mi[k] * Wh_w[(D_H + k) * D_H + tid];
        out_hi[(size_t)b * D_H + tid] = silu_f(acc);
    }
}

extern "C" void kernel_launch(void* const* d_in, const int* in_sizes, int n_in,
                              void* d_out, int out_size, void* d_ws, size_t ws_size,
                              hipStream_t stream) {
    const float* hi   = (const float*)d_in[0];
    const float* xi   = (const float*)d_in[1];
    const float* hj   = (const float*)d_in[2];
    const float* xj   = (const float*)d_in[3];
    const float* We_w = (const float*)d_in[4];
    const float* We_b = (const float*)d_in[5];
    const float* Wx_w = (const float*)d_in[6];
    const float* Wx_b = (const float*)d_in[7];
    const float* Wh_w = (const float*)d_in[8];
    const float* Wh_b = (const float*)d_in[9];

    float* out    = (float*)d_out;
    float* out_hi = out;
    float* out_xi = out_hi + (size_t)B_NODES * D_H;
    float* out_hj = out_xi + (size_t)B_NODES * D_X;
    float* out_xj = out_hj + (size_t)B_NODES * N_NBR * D_H;

    egnn_wmma_kernel<<<B_NODES, 256, 0, stream>>>(hi, xi, hj, xj,
                                                  We_w, We_b, Wx_w, Wx_b,
                                                  Wh_w, Wh_b,
                                                  out_hi, out_xi, out_hj, out_xj);
}